// GCN_55113020342885
// MI455X (gfx1250) — compile-verified
//
#include <hip/hip_runtime.h>

// GCN 2-layer forward for MI455X (gfx1250, wave32).
// GEMMs use V_WMMA_F32_16X16X4_F32 (full-precision fp32 matrix pipe).
// Edge aggregation uses float4 gathers + global fp32 atomics; both
// intermediate feature matrices fit in the 192MB L2, so the scatter-add
// phase is L2-resident and HBM traffic is ~200MB total.

typedef float v2f __attribute__((ext_vector_type(2)));
typedef float v8f __attribute__((ext_vector_type(8)));

#define N_NODES 100000
#define N_EDGES 1600000
#define F_IN    256
#define F_HID   64
#define F_OUT   40

// ---------------------------------------------------------------- utilities
__global__ void zero_kernel(float* __restrict__ p, int n) {
    int i = blockIdx.x * blockDim.x + threadIdx.x;
    if (i < n) p[i] = 0.0f;
}

__global__ void degree_kernel(const int* __restrict__ src, const int* __restrict__ dst,
                              float* __restrict__ outdeg, float* __restrict__ indeg) {
    int e = blockIdx.x * blockDim.x + threadIdx.x;
    if (e < N_EDGES) {
        atomicAdd(&outdeg[src[e]], 1.0f);
        atomicAdd(&indeg[dst[e]], 1.0f);
    }
}

__global__ void norm_kernel(const float* __restrict__ outdeg, const float* __restrict__ indeg,
                            float* __restrict__ ns, float* __restrict__ nd) {
    int i = blockIdx.x * blockDim.x + threadIdx.x;
    if (i < N_NODES) {
        ns[i] = rsqrtf(fmaxf(outdeg[i], 1.0f));
        nd[i] = rsqrtf(fmaxf(indeg[i], 1.0f));
    }
}

// ------------------------------------------------------------- layer-1 GEMM
// H1[100000 x 64] = diag(norm_src) * X[100000 x 256] @ W1[256 x 64]
// One wave per 16x16 C-tile; K stepped by 4 with fp32 WMMA.
__global__ void __launch_bounds__(256) gemm1_wmma(const float* __restrict__ X,
                                                  const float* __restrict__ W,
                                                  const float* __restrict__ ns,
                                                  float* __restrict__ H) {
    const int wave = (blockIdx.x * blockDim.x + threadIdx.x) >> 5;
    if (wave >= (N_NODES / 16) * (F_HID / 16)) return;   // wave-uniform guard
    const int tile_m = wave >> 2;        // 6250 row tiles
    const int tile_n = wave & 3;         // 4 col tiles
    const int lane = threadIdx.x & 31;
    const int half = lane >> 4;          // 0: K=0,1  1: K=2,3 (ISA 16x4 A layout)
    const int l16  = lane & 15;
    const int row  = tile_m * 16 + l16;  // A-fragment row (M)
    const int col  = tile_n * 16 + l16;  // B-fragment / C col (N)

    const float* xr = X + (size_t)row * F_IN + 2 * half;       // v2f loads
    const float* wc = W + (size_t)(2 * half) * F_HID + col;

    v8f c = {};
#pragma unroll 8
    for (int k = 0; k < F_IN; k += 4) {
        v2f a = *(const v2f*)(xr + k);
        v2f b;
        b.x = wc[(size_t)k * F_HID];         // W[k + 2*half][col]
        b.y = wc[(size_t)(k + 1) * F_HID];   // W[k+1+2*half][col]
        c = __builtin_amdgcn_wmma_f32_16x16x4_f32(false, a, false, b,
                                                  (short)0, c, false, false);
    }
    // row-scaling by norm_src hoisted out of the K-loop (per-row scalar)
#pragma unroll
    for (int v = 0; v < 8; ++v) {
        int r = tile_m * 16 + v + 8 * half;  // ISA C layout: VGPR v -> row v (+8 upper half)
        H[(size_t)r * F_HID + col] = c[v] * ns[r];
    }
}

// ------------------------------------------------------------- layer-2 GEMM
// H2[100000 x 40] = X2[100000 x 64] @ W2[64 x 40]   (X2 pre-scaled by norm_src)
// N=40 -> 3 tiles of 16 with zero-padded B and masked stores.
__global__ void __launch_bounds__(256) gemm2_wmma(const float* __restrict__ X2,
                                                  const float* __restrict__ W2,
                                                  float* __restrict__ H2) {
    const int wave = (blockIdx.x * blockDim.x + threadIdx.x) >> 5;
    if (wave >= (N_NODES / 16) * 3) return;
    const int tile_m = wave / 3;
    const int tile_n = wave % 3;
    const int lane = threadIdx.x & 31;
    const int half = lane >> 4;
    const int l16  = lane & 15;
    const int row  = tile_m * 16 + l16;
    const int col  = tile_n * 16 + l16;
    const bool colv = (col < F_OUT);

    const float* xr = X2 + (size_t)row * F_HID + 2 * half;

    v8f c = {};
#pragma unroll
    for (int k = 0; k < F_HID; k += 4) {
        v2f a = *(const v2f*)(xr + k);
        v2f b;
        b.x = colv ? W2[(size_t)(k + 2 * half) * F_OUT + col] : 0.0f;
        b.y = colv ? W2[(size_t)(k + 1 + 2 * half) * F_OUT + col] : 0.0f;
        c = __builtin_amdgcn_wmma_f32_16x16x4_f32(false, a, false, b,
                                                  (short)0, c, false, false);
    }
    if (colv) {
#pragma unroll
        for (int v = 0; v < 8; ++v) {
            int r = tile_m * 16 + v + 8 * half;
            H2[(size_t)r * F_OUT + col] = c[v];
        }
    }
}

// -------------------------------------------------------- edge scatter-add
// M[dst[e], :] += H[src[e], :]   float4 per thread; L2-resident working set.
template <int F>
__global__ void aggregate4_kernel(const float* __restrict__ H,
                                  const int* __restrict__ src,
                                  const int* __restrict__ dst,
                                  float* __restrict__ M) {
    constexpr int FQ = F / 4;
    int gid = blockIdx.x * blockDim.x + threadIdx.x;
    int e = gid / FQ;
    int q = gid - e * FQ;
    if (e < N_EDGES) {
        int s = src[e];
        int d = dst[e];
        const float4 h = *(const float4*)(H + (size_t)s * F + 4 * q);
        float* mp = M + (size_t)d * F + 4 * q;
        atomicAdd(mp + 0, h.x);
        atomicAdd(mp + 1, h.y);
        atomicAdd(mp + 2, h.z);
        atomicAdd(mp + 3, h.w);
    }
}

// ------------------------------------------------------------- epilogues
// h = M1*nd + b1  (-> output slot 1);  X2 = relu(h)*ns  (layer-2 input)
__global__ void finish1_kernel(const float* __restrict__ M1, const float* __restrict__ nd,
                               const float* __restrict__ ns, const float* __restrict__ b1,
                               float* __restrict__ out_h, float* __restrict__ X2) {
    int i = blockIdx.x * blockDim.x + threadIdx.x;
    if (i < N_NODES * F_HID) {
        int n = i >> 6;
        int f = i & (F_HID - 1);
        float h = fmaf(M1[i], nd[n], b1[f]);
        out_h[i] = h;
        X2[i] = fmaxf(h, 0.0f) * ns[n];
    }
}

// x = M2*nd + b2  (-> output slot 0)
__global__ void finish2_kernel(const float* __restrict__ M2, const float* __restrict__ nd,
                               const float* __restrict__ b2, float* __restrict__ out_x) {
    int i = blockIdx.x * blockDim.x + threadIdx.x;
    if (i < N_NODES * F_OUT) {
        int n = i / F_OUT;
        int f = i - n * F_OUT;
        out_x[i] = fmaf(M2[i], nd[n], b2[f]);
    }
}

// ---------------------------------------------------------------- launcher
extern "C" void kernel_launch(void* const* d_in, const int* in_sizes, int n_in,
                              void* d_out, int out_size, void* d_ws, size_t ws_size,
                              hipStream_t stream) {
    const float* X  = (const float*)d_in[0];
    const int*   src = (const int*)d_in[1];   // integer inputs -> const int* per harness
    const int*   dst = (const int*)d_in[2];
    const float* W1 = (const float*)d_in[3];
    const float* b1 = (const float*)d_in[4];
    const float* W2 = (const float*)d_in[5];
    const float* b2 = (const float*)d_in[6];
    float* out = (float*)d_out;               // [x: 100000*40][h: 100000*64]
    float* ws  = (float*)d_ws;

    // workspace layout (floats); bufA/bufB reused across phases
    float* outdeg = ws;                          // 100000
    float* indeg  = ws + N_NODES;                // 100000
    float* ns     = ws + 2 * N_NODES;            // 100000
    float* nd     = ws + 3 * N_NODES;            // 100000
    float* bufA   = ws + 4 * N_NODES;            // 6,400,000: H1 -> X2 -> M2
    float* bufB   = bufA + (size_t)N_NODES * F_HID; // 6,400,000: M1 -> H2

    const int T = 256;

    // zero atomic accumulators (degrees + M1)
    zero_kernel<<<(2 * N_NODES + T - 1) / T, T, 0, stream>>>(ws, 2 * N_NODES);
    zero_kernel<<<(N_NODES * F_HID + T - 1) / T, T, 0, stream>>>(bufB, N_NODES * F_HID);

    degree_kernel<<<(N_EDGES + T - 1) / T, T, 0, stream>>>(src, dst, outdeg, indeg);
    norm_kernel<<<(N_NODES + T - 1) / T, T, 0, stream>>>(outdeg, indeg, ns, nd);

    // Layer 1: WMMA GEMM -> scatter-add -> epilogue
    {
        int waves = (N_NODES / 16) * (F_HID / 16);          // 25000
        gemm1_wmma<<<(waves * 32 + T - 1) / T, T, 0, stream>>>(X, W1, ns, bufA);
    }
    aggregate4_kernel<F_HID><<<(N_EDGES * (F_HID / 4) + T - 1) / T, T, 0, stream>>>(bufA, src, dst, bufB);
    finish1_kernel<<<(N_NODES * F_HID + T - 1) / T, T, 0, stream>>>(
        bufB, nd, ns, b1, out + (size_t)N_NODES * F_OUT, bufA);

    // Layer 2: WMMA GEMM -> zero M2 -> scatter-add -> epilogue
    {
        int waves = (N_NODES / 16) * 3;                     // 18750
        gemm2_wmma<<<(waves * 32 + T - 1) / T, T, 0, stream>>>(bufA, W2, bufB);
    }
    zero_kernel<<<(N_NODES * F_OUT + T - 1) / T, T, 0, stream>>>(bufA, N_NODES * F_OUT);
    aggregate4_kernel<F_OUT><<<(N_EDGES * (F_OUT / 4) + T - 1) / T, T, 0, stream>>>(bufB, src, dst, bufA);
    finish2_kernel<<<(N_NODES * F_OUT + T - 1) / T, T, 0, stream>>>(bufA, nd, b2, out);
}